// SIBlocks_49185965474359
// MI455X (gfx1250) — compile-verified
//
#include <hip/hip_runtime.h>
#include <math.h>

typedef __attribute__((ext_vector_type(2))) float v2f;
typedef __attribute__((ext_vector_type(8))) float v8f;

#define GRID_H 48
#define NPTS   (48 * 48)       // 2304
#define CCH    16
#define NKNOT  16
#define KITERS (NPTS / 16)     // 144 — compile-time trip count (uniform EXEC)
#define RADIUS_C 0.2f
#define EPS_H_C  1e-6f
#define EPS_T_C  1e-8f

// One block = 4 waves = 128 threads. Block handles 16 output rows (i-tile).
// Each wave takes a strided quarter of the j (K) range; weights for a 16x4
// K-slab are computed in registers laid out exactly as the f32 WMMA A operand,
// then two V_WMMA_F32_16X16X4_F32 (one per batch) accumulate out tiles.
__global__ __launch_bounds__(128) void si_fused_wmma_kernel(
    const float* __restrict__ x,
    const float* __restrict__ phi_w1, const float* __restrict__ phi_b1,
    const float* __restrict__ phi_w2, const float* __restrict__ phi_b2,
    const float* __restrict__ h_w1,  const float* __restrict__ h_b1,
    const float* __restrict__ h_w2,  const float* __restrict__ h_b2,
    const float* __restrict__ S_m,
    float* __restrict__ out)
{
  __shared__ float S_s[NKNOT];
  __shared__ float accs[4][2][8][32];   // [wave][batch][vgpr r][lane]
  __shared__ float cnts[4][16];         // [wave][row-in-tile] mask counts

  const int tid  = threadIdx.x;
  const int wave = tid >> 5;
  const int lane = tid & 31;
  const int li   = lane & 15;           // row-in-tile (A) / column c (B,D)
  const int half = lane >> 4;           // 0: K=0,1   1: K=2,3
  const int i0   = blockIdx.x * 16;

  if (tid < NKNOT) S_s[tid] = S_m[tid];
  __syncthreads();

  // ---- per-row (i) setup: coords, bandwidth h_i, phi row-partials ----
  const int row = i0 + li;
  const int ai  = row / GRID_H;
  const int bi  = row - ai * GRID_H;
  const float cx = (float)ai * (1.0f / (GRID_H - 1));
  const float cy = (float)bi * (1.0f / (GRID_H - 1));

  float hz = h_b2[0];
#pragma unroll
  for (int k = 0; k < 32; ++k) {
    float t = fmaf(cx, h_w1[k], fmaf(cy, h_w1[32 + k], h_b1[k]));
    t = fmaxf(t, 0.0f);
    hz = fmaf(t, h_w2[k], hz);
  }
  const float hval  = (hz > 30.0f) ? hz : log1pf(__expf(hz)); // softplus
  const float inv_h = 1.0f / (hval + EPS_H_C);

  float Ai[32], Bx[32], By[32], W2[32];
#pragma unroll
  for (int k = 0; k < 32; ++k) {
    Ai[k] = fmaf(cx, phi_w1[k], fmaf(cy, phi_w1[32 + k], phi_b1[k]));
    Bx[k] = phi_w1[64 + k];
    By[k] = phi_w1[96 + k];
    W2[k] = phi_w2[k];
  }
  const float pb2 = phi_b2[0];

  v8f acc0 = {};
  v8f acc1 = {};
  float cnt = 0.0f;

  // ---- K loop: constant trip count so EXEC stays structurally all-ones ----
  for (int t = 0; t < KITERS; ++t) {
    const int j0 = wave * 4 + t * 16;
    const int jj = j0 + half * 2;       // this lane's first K column
    v2f a, bop0, bop1;
#pragma unroll
    for (int u = 0; u < 2; ++u) {
      const int j  = jj + u;
      const int aj = j / GRID_H;
      const int bj = j - aj * GRID_H;
      const float cxj = (float)aj * (1.0f / (GRID_H - 1));
      const float cyj = (float)bj * (1.0f / (GRID_H - 1));

      const float dx = cx - cxj, dy = cy - cyj;
      const float dist = sqrtf(fmaf(dx, dx, dy * dy));
      const float msk  = (dist <= RADIUS_C) ? 1.0f : 0.0f;

      // radial spline on r = dist / (h_i + eps), 16 uniform knots
      float r = dist * inv_h;
      r = fminf(fmaxf(r, 0.0f), 1.0f);
      int idx = (int)(r * (float)(NKNOT - 1));
      idx = (idx < 0) ? 0 : ((idx > NKNOT - 2) ? NKNOT - 2 : idx);
      const float tk  = (float)idx * (1.0f / (NKNOT - 1));
      const float wr  = (r - tk) * (1.0f / ((1.0f / (NKNOT - 1)) + EPS_T_C));
      const float s0v = S_s[idx];
      const float psi = fmaf(wr, S_s[idx + 1] - s0v, s0v);

      // phi MLP: hidden = relu(Ai + cxj*Bx + cyj*By); phi = hidden . W2 + b2
      float phi = pb2;
#pragma unroll
      for (int k = 0; k < 32; ++k) {
        float hd = fmaf(cxj, Bx[k], fmaf(cyj, By[k], Ai[k]));
        hd = fmaxf(hd, 0.0f);
        phi = fmaf(hd, W2[k], phi);
      }

      a[u] = phi * psi * msk;           // weight w[row, j]
      cnt += msk;

      // B operand: x[b, j, c=li]
      bop0[u] = x[j * CCH + li];
      bop1[u] = x[NPTS * CCH + j * CCH + li];
    }

    // D(16x16,f32) += A(16x4,f32) x B(4x16,f32)  -- exact fp32 math
    acc0 = __builtin_amdgcn_wmma_f32_16x16x4_f32(
        false, a, false, bop0, (short)0, acc0, false, false);
    acc1 = __builtin_amdgcn_wmma_f32_16x16x4_f32(
        false, a, false, bop1, (short)0, acc1, false, false);
  }

  // lanes L and L+16 hold mask counts for the same row over disjoint j's
  const float cnt_row = cnt + __shfl_xor(cnt, 16, 32);
  if (half == 0) cnts[wave][li] = cnt_row;
#pragma unroll
  for (int r = 0; r < 8; ++r) {
    accs[wave][0][r][lane] = acc0[r];
    accs[wave][1][r][lane] = acc1[r];
  }
  __syncthreads();

  // ---- cross-wave reduction + normalization + store (wave 0) ----
  if (wave == 0) {
#pragma unroll
    for (int r = 0; r < 8; ++r) {
      const int rrow = r + (half << 3);      // D layout: rows r / r+8 per half
      const float nrm = fmaxf(cnts[0][rrow] + cnts[1][rrow] +
                              cnts[2][rrow] + cnts[3][rrow], 1.0f);
      const float inv = 1.0f / nrm;
      const float v0 = accs[0][0][r][lane] + accs[1][0][r][lane] +
                       accs[2][0][r][lane] + accs[3][0][r][lane];
      const float v1 = accs[0][1][r][lane] + accs[1][1][r][lane] +
                       accs[2][1][r][lane] + accs[3][1][r][lane];
      out[(i0 + rrow) * CCH + li] = v0 * inv;
      out[NPTS * CCH + (i0 + rrow) * CCH + li] = v1 * inv;
    }
  }
}

extern "C" void kernel_launch(void* const* d_in, const int* in_sizes, int n_in,
                              void* d_out, int out_size, void* d_ws, size_t ws_size,
                              hipStream_t stream) {
  const float* x      = (const float*)d_in[0];
  const float* phi_w1 = (const float*)d_in[1];
  const float* phi_b1 = (const float*)d_in[2];
  const float* phi_w2 = (const float*)d_in[3];
  const float* phi_b2 = (const float*)d_in[4];
  const float* h_w1   = (const float*)d_in[5];
  const float* h_b1   = (const float*)d_in[6];
  const float* h_w2   = (const float*)d_in[7];
  const float* h_b2   = (const float*)d_in[8];
  const float* S_m    = (const float*)d_in[9];
  float* out = (float*)d_out;

  dim3 grid(NPTS / 16);   // 144 row tiles
  dim3 block(128);        // 4 waves
  si_fused_wmma_kernel<<<grid, block, 0, stream>>>(
      x, phi_w1, phi_b1, phi_w2, phi_b2,
      h_w1, h_b1, h_w2, h_b2, S_m, out);
}